// Model_3152505996047
// MI455X (gfx1250) — compile-verified
//
#include <hip/hip_runtime.h>

typedef __attribute__((ext_vector_type(2))) float v2f;
typedef __attribute__((ext_vector_type(8))) float v8f;

#define IN_FEATS  128
#define OUT_FEATS 64

// Native no-return f32 atomic add, device scope, SADDR form:
//   global_atomic_add_f32 v_voff, v_data, s[base:base+1] offset:<imm>
#define ATOMIC_ADD_F32_NORET(basePtr, voff, val, offlit)                        \
    asm volatile("global_atomic_add_f32 %0, %1, %2 offset:" #offlit            \
                 " scope:SCOPE_DEV"                                            \
                 :: "v"(voff), "v"(val), "s"(basePtr) : "memory")

// ---------------------------------------------------------------------------
// Kernel 0: zero the output accumulator (harness poisons d_out with 0xAA).
// ---------------------------------------------------------------------------
__global__ void __launch_bounds__(256) zero_f4(float4* __restrict__ p, int n4) {
    int i = blockIdx.x * blockDim.x + threadIdx.x;
    if (i < n4) p[i] = make_float4(0.f, 0.f, 0.f, 0.f);
}

// ---------------------------------------------------------------------------
// Kernel 1: h = feat @ W^T + b   via V_WMMA_F32_16X16X4_F32.
// One wave computes one 16x16 tile; K=128 in 32 steps of 4, split into two
// independent accumulator chains (even/odd K-steps) to hide WMMA D->C latency.
//
// A (16x4, MxK) per-lane layout (ISA 7.12.2):
//   lanes 0-15 : M=lane,    v0=K(k0),   v1=K(k0+1)
//   lanes 16-31: M=lane-16, v0=K(k0+2), v1=K(k0+3)
// B (4x16, KxN): mirrored — lane selects column N=lane%16, halves select K.
// B[k][n] = W[n][k]  (W stored [OUT,IN] row-major).
// C/D (16x16 f32): VGPR i, lane l -> row = i + 8*(l>>4), col = l&15.
// ---------------------------------------------------------------------------
__global__ void __launch_bounds__(256) gemm_bias_wmma(
    const float* __restrict__ feat, const float* __restrict__ W,
    const float* __restrict__ bias, float* __restrict__ h, int n_nodes) {
    const int lane   = threadIdx.x & 31;
    const int wave   = blockIdx.x * (blockDim.x >> 5) + (threadIdx.x >> 5);
    const int tilesN = OUT_FEATS / 16;             // 4
    const int tilesM = n_nodes / 16;               // 3125
    if (wave >= tilesM * tilesN) return;           // wave-uniform: EXEC stays full
    const int mt = wave / tilesN;
    const int nt = wave % tilesN;

    const int half = lane >> 4;                    // 0 or 1 (K sub-pair)
    const int l16  = lane & 15;

    const float* __restrict__ arow = feat + (size_t)(mt * 16 + l16) * IN_FEATS;
    const float* __restrict__ brow = W    + (size_t)(nt * 16 + l16) * IN_FEATS;

    v8f c0 = {}, c1 = {};
#pragma unroll
    for (int k0 = 0; k0 < IN_FEATS; k0 += 8) {
        const int ka = k0 + 2 * half;
        const int kb = k0 + 4 + 2 * half;
        v2f a0, b0, a1, b1;
        a0.x = arow[ka]; a0.y = arow[ka + 1];
        b0.x = brow[ka]; b0.y = brow[ka + 1];
        a1.x = arow[kb]; a1.y = arow[kb + 1];
        b1.x = brow[kb]; b1.y = brow[kb + 1];
        // 8 args: (neg_a, A, neg_b, B, c_mod, C, reuse_a, reuse_b)
        c0 = __builtin_amdgcn_wmma_f32_16x16x4_f32(
                 false, a0, false, b0, (short)0, c0, false, false);
        c1 = __builtin_amdgcn_wmma_f32_16x16x4_f32(
                 false, a1, false, b1, (short)0, c1, false, false);
    }

    const float bv = bias[nt * 16 + l16];          // bias depends on column only
    float* __restrict__ ho = h + (size_t)(mt * 16) * OUT_FEATS + nt * 16 + l16;
#pragma unroll
    for (int i = 0; i < 8; ++i)
        ho[(size_t)(i + 8 * half) * OUT_FEATS] = c0[i] + c1[i] + bv;
}

// ---------------------------------------------------------------------------
// Kernel 2: out[dst] += h[src] * w[e].
// Two edges per wave: 16 lanes per edge, one b128 gather (256B/edge-row) and
// four no-return global_atomic_add_f32 per lane. h (12.8 MB) and out (12.8 MB)
// are L2-resident; edge streams are single-use -> nontemporal loads.
// ---------------------------------------------------------------------------
__global__ void __launch_bounds__(256) edge_scatter(
    const int* __restrict__ esrc, const int* __restrict__ edst,
    const float* __restrict__ ew, const float* __restrict__ h,
    float* __restrict__ out, int n_edges) {
    const int lane = threadIdx.x & 31;
    const int g    = lane >> 4;                    // which edge of the pair
    const int sub  = lane & 15;                    // 16 lanes x float4 per edge
    const int wpb  = blockDim.x >> 5;
    int       pair = blockIdx.x * wpb + (threadIdx.x >> 5);
    const int strd = gridDim.x * wpb;
    const int npairs = (n_edges + 1) >> 1;

    for (int p = pair; p < npairs; p += strd) {
        const int e = 2 * p + g;
        if (e < n_edges) {
            const int   s = __builtin_nontemporal_load(esrc + e);
            const int   d = __builtin_nontemporal_load(edst + e);
            const float w = __builtin_nontemporal_load(ew + e);
            const float4 hv = ((const float4*)(h + (size_t)s * OUT_FEATS))[sub];
            const unsigned voff =
                ((unsigned)d * OUT_FEATS + (unsigned)sub * 4u) * sizeof(float);
            ATOMIC_ADD_F32_NORET(out, voff, hv.x * w, 0);
            ATOMIC_ADD_F32_NORET(out, voff, hv.y * w, 4);
            ATOMIC_ADD_F32_NORET(out, voff, hv.z * w, 8);
            ATOMIC_ADD_F32_NORET(out, voff, hv.w * w, 12);
        }
    }
}

// ---------------------------------------------------------------------------
// Launch: inputs are (feat, edge_src, edge_dst, eweight, W, b).
// d_ws holds h [n_nodes x OUT_FEATS] fp32 (12.8 MB).
// ---------------------------------------------------------------------------
extern "C" void kernel_launch(void* const* d_in, const int* in_sizes, int n_in,
                              void* d_out, int out_size, void* d_ws, size_t ws_size,
                              hipStream_t stream) {
    const float* feat = (const float*)d_in[0];
    const int*   esrc = (const int*)  d_in[1];
    const int*   edst = (const int*)  d_in[2];
    const float* ew   = (const float*)d_in[3];
    const float* W    = (const float*)d_in[4];
    const float* b    = (const float*)d_in[5];
    float*       out  = (float*)d_out;
    float*       h    = (float*)d_ws;

    const int out_feats = in_sizes[5];                  // 64
    const int in_feats  = in_sizes[4] / out_feats;      // 128
    const int n_nodes   = in_sizes[0] / in_feats;       // 50000
    const int n_edges   = in_sizes[1];                  // 800000

    // 1) zero output accumulator
    const int n4 = out_size / 4;
    zero_f4<<<(n4 + 255) / 256, 256, 0, stream>>>((float4*)out, n4);

    // 2) WMMA GEMM + bias into workspace
    const int tiles = (n_nodes / 16) * (OUT_FEATS / 16);
    const int wpb   = 256 / 32;                         // 8 waves per block
    gemm_bias_wmma<<<(tiles + wpb - 1) / wpb, 256, 0, stream>>>(
        feat, W, b, h, n_nodes);

    // 3) edge gather / scale / scatter-add (2 edges per wave)
    edge_scatter<<<4096, 256, 0, stream>>>(esrc, edst, ew, h, out, n_edges);
}